// NONA_NN_15169824489968
// MI455X (gfx1250) — compile-verified
//
#include <hip/hip_runtime.h>
#include <math.h>

// ---------------------------------------------------------------------------
// NONA head, fused flash-attention style for MI455X (gfx1250, wave32, WMMA).
// softmax(max(dist)-dist) == softmax(-dist) (shift invariance), and since
// s = -dist <= 0, exp(s) cannot overflow -> no online max needed at all.
// Row sums are produced by the P@Y WMMA via an appended ones-column in Y.
// ---------------------------------------------------------------------------

typedef __attribute__((ext_vector_type(2))) float v2f;
typedef __attribute__((ext_vector_type(8))) float v8f;

#define NN        8192
#define DIN       256
#define H0        128
#define DH        64
#define NCLS      10
#define SUMCOL    10                         // ones-column -> row sum of P
#define CCHUNK    4
#define COLS_PER_CHUNK (NN / CCHUNK)         // 2048
#define COLTILE   128
#define NSTEPS    (COLS_PER_CHUNK / COLTILE) // 16

// raw v_sqrt_f32: inputs here are in [0, ~1e3], no denorm/overflow fixup needed
__device__ __forceinline__ float fast_sqrtf(float x) {
#if __has_builtin(__builtin_amdgcn_sqrtf)
    return __builtin_amdgcn_sqrtf(x);
#else
    return sqrtf(x);
#endif
}

// ---------------- small MLP (two linear layers, no activation) -------------
__global__ __launch_bounds__(256) void nona_mlp(
    const float* __restrict__ in, const float* __restrict__ W,
    const float* __restrict__ b, float* __restrict__ out,
    int nrows, int din, int dout)
{
    int gid = blockIdx.x * 256 + threadIdx.x;
    if (gid >= nrows * dout) return;
    int row = gid / dout, o = gid % dout;
    const float4* ip = (const float4*)(in + (size_t)row * din);
    const float4* wp = (const float4*)(W + (size_t)o * din);
    float acc = b[o];
    int k4 = din >> 2;
    for (int k = 0; k < k4; ++k) {
        float4 a = ip[k], w = wp[k];
        acc += a.x * w.x + a.y * w.y + a.z * w.z + a.w * w.w;
    }
    out[gid] = acc;
}

// ---------------- row squared-norms of h [NN x DH] -------------------------
__global__ __launch_bounds__(256) void nona_sq(
    const float* __restrict__ h, float* __restrict__ sq)
{
    int row = blockIdx.x * 256 + threadIdx.x;
    if (row >= NN) return;
    const float4* p = (const float4*)(h + (size_t)row * DH);
    float s = 0.f;
#pragma unroll
    for (int k = 0; k < DH / 4; ++k) {
        float4 a = p[k];
        s += a.x * a.x + a.y * a.y + a.z * a.z + a.w * a.w;
    }
    sq[row] = s;
}

// ---------------- fused dist -> exp -> P@[Y|1] -----------------------------
// Workgroup: 256 threads = 8 waves; wave w owns rows [rowBase, rowBase+16).
// Grid: (NN/128, CCHUNK). Each block handles one 2048-column chunk.
__global__ __launch_bounds__(256) void nona_attn(
    const float* __restrict__ h,   const float* __restrict__ hn,
    const float* __restrict__ sqx, const float* __restrict__ sqn,
    const float* __restrict__ y,
    float* __restrict__ pacc)
{
    __shared__ float ldsHn[COLTILE * DH];   // 32 KB neighbor tile
    __shared__ float ldsY [COLTILE * 16];   // 8 KB  Y tile (col 10 = ones)
    __shared__ float ldsSq[COLTILE];        // column sq-norms for this step
    __shared__ float ldsP [8][16 * 16];     // 8 KB  per-wave P transpose scratch

    const int tid  = threadIdx.x;
    const int w    = tid >> 5;
    const int lane = tid & 31;
    const int n16  = lane & 15;      // column-within-tile / row-within-A
    const int hi   = lane >> 4;      // half select
    const int rowBase  = blockIdx.x * 128 + w * 16;
    const int colStart = blockIdx.y * COLS_PER_CHUNK;

    // A operand (16x64 row strip) is invariant over the whole column sweep:
    // A layout (f32 16x4): lanes 0-15: M=lane, K={0,1}; lanes 16-31: M=lane-16, K={2,3}
    v2f Areg[16];
    {
        const int arow = rowBase + n16;
#pragma unroll
        for (int kk = 0; kk < 16; ++kk)
            Areg[kk] = *(const v2f*)(h + (size_t)arow * DH + 4 * kk + 2 * hi);
    }

    float sqm[8];
#pragma unroll
    for (int v = 0; v < 8; ++v) sqm[v] = sqx[rowBase + v + 8 * hi];

    v8f acc;   // C/D layout: acc[v] @ lane = out[row v+8*hi][col n16]
#pragma unroll
    for (int v = 0; v < 8; ++v) acc[v] = 0.f;

    for (int t = 0; t < NSTEPS; ++t) {
        const int colBase = colStart + t * COLTILE;
        __syncthreads();
        // cooperative stage: 128x64 neighbor tile (2048 float4)
#pragma unroll
        for (int i = 0; i < 8; ++i) {
            int idx = tid + i * 256;
            int r = idx >> 4, c4 = idx & 15;
            *(float4*)&ldsHn[r * DH + c4 * 4] =
                *(const float4*)(hn + (size_t)(colBase + r) * DH + c4 * 4);
        }
        // cooperative stage: 128x16 Y tile; col 10 = 1.0 (row-sum column)
#pragma unroll
        for (int i = 0; i < 8; ++i) {
            int idx = tid + i * 256;
            int r = idx >> 4, c = idx & 15;
            ldsY[idx] = (c < NCLS) ? y[(size_t)(colBase + r) * NCLS + c]
                                   : (c == SUMCOL ? 1.0f : 0.f);
        }
        // column sq-norms
        if (tid < COLTILE) ldsSq[tid] = sqn[colBase + tid];
        __syncthreads();

        for (int j = 0; j < 8; ++j) {
            const int colB = colBase + j * 16;
            const float sqn_l = ldsSq[j * 16 + n16];

            // ---- Gram 16x16 tile via f32 WMMA, K=64 in 16 steps of 4 ----
            v8f cg;
#pragma unroll
            for (int v = 0; v < 8; ++v) cg[v] = 0.f;
#pragma unroll
            for (int kk = 0; kk < 16; ++kk) {
                // B layout (f32 4x16): VGPR0 = B[2*hi][n], VGPR1 = B[2*hi+1][n]
                v2f b = *(const v2f*)&ldsHn[(j * 16 + n16) * DH + 4 * kk + 2 * hi];
                cg = __builtin_amdgcn_wmma_f32_16x16x4_f32(
                        false, Areg[kk], false, b, (short)0, cg, false, false);
            }

            // ---- p = exp(-dist); diagonal tile only when rowBase == colB ----
            if (rowBase == colB) {          // wave-uniform (both 16-aligned)
#pragma unroll
                for (int v = 0; v < 8; ++v) {
                    float d2 = fmaxf(sqm[v] + sqn_l - 2.0f * cg[v], 0.f);
                    float p  = __expf(-fast_sqrtf(d2));
                    if (n16 == v + 8 * hi) p = 0.f;   // mask self-similarity
                    ldsP[w][(v + 8 * hi) * 16 + n16] = p;
                }
            } else {
#pragma unroll
                for (int v = 0; v < 8; ++v) {
                    float d2 = fmaxf(sqm[v] + sqn_l - 2.0f * cg[v], 0.f);
                    ldsP[w][(v + 8 * hi) * 16 + n16] = __expf(-fast_sqrtf(d2));
                }
            }
            __builtin_amdgcn_fence(__ATOMIC_ACQ_REL, "wavefront");

            // ---- acc += P(16x16) @ [Y|1](16x16) via f32 WMMA, K=16 ----
#pragma unroll
            for (int kk = 0; kk < 4; ++kk) {
                v2f a2 = *(const v2f*)&ldsP[w][n16 * 16 + 4 * kk + 2 * hi];
                v2f b2;
                b2.x = ldsY[(j * 16 + 4 * kk + 2 * hi    ) * 16 + n16];
                b2.y = ldsY[(j * 16 + 4 * kk + 2 * hi + 1) * 16 + n16];
                acc = __builtin_amdgcn_wmma_f32_16x16x4_f32(
                        false, a2, false, b2, (short)0, acc, false, false);
            }
        }
    }

    // ---- spill per-chunk partial acc (cols 0..9 = y-sums, col 10 = l) ----
#pragma unroll
    for (int v = 0; v < 8; ++v) {
        const int grow = rowBase + v + 8 * hi;
        pacc[((size_t)grow * CCHUNK + blockIdx.y) * 16 + n16] = acc[v];
    }
}

// ---------------- merge the CCHUNK partial sums and normalize --------------
__global__ __launch_bounds__(256) void nona_merge(
    const float* __restrict__ pacc, float* __restrict__ out)
{
    int gid = blockIdx.x * 256 + threadIdx.x;
    if (gid >= NN * NCLS) return;
    int row = gid / NCLS, c = gid % NCLS;
    float A = 0.f, L = 0.f;
#pragma unroll
    for (int k = 0; k < CCHUNK; ++k) {
        const float* p = pacc + ((size_t)row * CCHUNK + k) * 16;
        A += p[c];
        L += p[SUMCOL];
    }
    out[gid] = A / L;
}

// ---------------------------------------------------------------------------
extern "C" void kernel_launch(void* const* d_in, const int* in_sizes, int n_in,
                              void* d_out, int out_size, void* d_ws, size_t ws_size,
                              hipStream_t stream)
{
    const float* x  = (const float*)d_in[0];
    const float* xn = (const float*)d_in[1];
    const float* y  = (const float*)d_in[2];
    const float* W0 = (const float*)d_in[3];
    const float* b0 = (const float*)d_in[4];
    const float* W1 = (const float*)d_in[5];
    const float* b1 = (const float*)d_in[6];
    float* out = (float*)d_out;
    float* ws  = (float*)d_ws;

    // workspace layout (floats)
    float* h0x  = ws;                         // 8192*128
    float* h1x  = h0x  + (size_t)NN * H0;     // 8192*64
    float* h0n  = h1x  + (size_t)NN * DH;     // 8192*128
    float* h1n  = h0n  + (size_t)NN * H0;     // 8192*64
    float* sqx  = h1n  + (size_t)NN * DH;     // 8192
    float* sqn  = sqx  + NN;                  // 8192
    float* pacc = sqn  + NN;                  // 8192*4*16

    // MLP trunk (both branches; x_neighbors == x in the reference setup)
    nona_mlp<<<(NN * H0) / 256, 256, 0, stream>>>(x,   W0, b0, h0x, NN, DIN, H0);
    nona_mlp<<<(NN * H0) / 256, 256, 0, stream>>>(xn,  W0, b0, h0n, NN, DIN, H0);
    nona_mlp<<<(NN * DH) / 256, 256, 0, stream>>>(h0x, W1, b1, h1x, NN, H0, DH);
    nona_mlp<<<(NN * DH) / 256, 256, 0, stream>>>(h0n, W1, b1, h1n, NN, H0, DH);
    nona_sq<<<NN / 256, 256, 0, stream>>>(h1x, sqx);
    nona_sq<<<NN / 256, 256, 0, stream>>>(h1n, sqn);

    // fused NONA head
    nona_attn<<<dim3(NN / 128, CCHUNK), 256, 0, stream>>>(
        h1x, h1n, sqx, sqn, y, pacc);

    // sum column chunks and normalize
    nona_merge<<<(NN * NCLS + 255) / 256, 256, 0, stream>>>(pacc, out);
}